// INGP_33243046871810
// MI455X (gfx1250) — compile-verified
//
#include <hip/hip_runtime.h>
#include <cmath>

typedef __attribute__((ext_vector_type(2))) float v2f;
typedef __attribute__((ext_vector_type(8))) float v8f;

#define NLEV 12

struct Level { unsigned off, hsize, r1, hashed; float scale; };
struct Levels { Level L[NLEV]; };

// ---- host: replicate _level_specs() exactly (float64 math, int64 dense) ----
static Levels make_levels(int D) {
  Levels out{};
  const double f = pow(8192.0 / 16.0, 1.0 / (NLEV - 1));
  long long offset = 0;
  for (int l = 0; l < NLEV; ++l) {
    double scale = 16.0 * pow(f, (double)l) - 1.0;
    long long res = (long long)ceil(scale) + 1;
    long long r1  = res + 1;
    long long dense = 1;
    for (int d = 0; d < D; ++d) dense *= r1;
    long long params = dense < (1ll << 21) ? dense : (1ll << 21);
    params = ((params + 7) / 8) * 8;
    out.L[l].off    = (unsigned)offset;
    out.L[l].hsize  = (unsigned)params;     // hashed levels: exactly 2^21 (pow2 -> AND mask)
    out.L[l].r1     = (unsigned)r1;
    out.L[l].hashed = (dense > (1ll << 21)) ? 1u : 0u;
    out.L[l].scale  = (float)scale;
    offset += params;
  }
  return out;
}

// ---- device: hash-grid encode of one point into a 24-float LDS row ----
// Per level: compute all 2^D indices/weights first, then issue all gathers
// (distinct destination registers -> loads pipeline, single progressive wait),
// then accumulate.
template <int D>
__device__ __forceinline__ void encode_point(const float* x01,
                                             const float2* __restrict__ table,
                                             const Levels& specs,
                                             float* __restrict__ ldsrow) {
  const unsigned PR[3] = {1u, 2654435761u, 805459861u};
  constexpr int NC = 1 << D;
#pragma unroll
  for (int l = 0; l < NLEV; ++l) {
    const Level sp = specs.L[l];
    float    wt[D][2];
    unsigned th[D][2];   // hash terms  coord*prime
    unsigned td[D][2];   // dense terms coord*stride
    unsigned stride = 1u;
#pragma unroll
    for (int d = 0; d < D; ++d) {
      float pos = x01[d] * sp.scale + 0.5f;   // align_corners=False
      float pf  = floorf(pos);
      float fr  = pos - pf;
      unsigned pg = (unsigned)pf;             // pos > 0 always
      wt[d][0] = 1.f - fr;       wt[d][1] = fr;
      th[d][0] = pg * PR[d];     th[d][1] = (pg + 1u) * PR[d];
      td[d][0] = pg * stride;    td[d][1] = (pg + 1u) * stride;
      stride *= sp.r1;
    }
    const unsigned mask = sp.hsize - 1u;      // hashed -> hsize == 2^21
    unsigned gidx[NC];
    float    w[NC];
#pragma unroll
    for (int c = 0; c < NC; ++c) {
      float    ww = wt[0][c & 1];
      unsigned ih = th[0][c & 1];
      unsigned id = td[0][c & 1];
#pragma unroll
      for (int d = 1; d < D; ++d) {
        const int bit = (c >> d) & 1;
        ww *= wt[d][bit];
        ih ^= th[d][bit];                     // uint32 wraparound xor-hash
        id += td[d][bit];                     // dense: < hsize by construction
      }
      gidx[c] = sp.off + (sp.hashed ? (ih & mask) : id);
      w[c] = ww;
    }
    float2 t[NC];
#pragma unroll
    for (int c = 0; c < NC; ++c) t[c] = table[gidx[c]];   // NC loads in flight
    float f0 = 0.f, f1 = 0.f;
#pragma unroll
    for (int c = 0; c < NC; ++c) {
      f0 = fmaf(w[c], t[c].x, f0);
      f1 = fmaf(w[c], t[c].y, f1);
    }
    ldsrow[2 * l]     = f0;
    ldsrow[2 * l + 1] = f1;
  }
}

// ---- device: MLP weights resident in registers as WMMA B-fragments ----
// W1 [24,64]: B-frag layout for V_WMMA_F32_16X16X4_F32 (B is 4x16):
//   lane L holds column n = nt*16 + (L&15); VGPR0 = K row kk0, VGPR1 = row kk0+1,
//   kk0 = 0 for lanes 0-15, 2 for lanes 16-31.
struct MlpW {
  v2f   Bf[6][4];
  float b1v[4], w2v[4], b2s;
};

__device__ __forceinline__ MlpW load_mlp(const float* __restrict__ w1,
                                         const float* __restrict__ b1,
                                         const float* __restrict__ w2,
                                         const float* __restrict__ b2, int lane) {
  MlpW W;
  const int l16 = lane & 15;
  const int kk0 = (lane >= 16) ? 2 : 0;
#pragma unroll
  for (int kt = 0; kt < 6; ++kt) {
#pragma unroll
    for (int nt = 0; nt < 4; ++nt) {
      int n = nt * 16 + l16;
      v2f v;
      v.x = w1[(kt * 4 + kk0) * 64 + n];
      v.y = w1[(kt * 4 + kk0 + 1) * 64 + n];
      W.Bf[kt][nt] = v;
    }
  }
#pragma unroll
  for (int nt = 0; nt < 4; ++nt) {
    W.b1v[nt] = b1[nt * 16 + l16];
    W.w2v[nt] = w2[nt * 16 + l16];
  }
  W.b2s = b2[0];
  return W;
}

// ---- device: 16-row tile of the MLP via WMMA f32 16x16x4 ----
// Preloads all six A fragments (batched ds loads, one wait), then 24 WMMAs as
// four independent accumulation chains. Returns o[m] + b2, valid for lanes with
// (lane&15)<8, row m = (lane>=16 ? 8 : 0) + (lane&7) of this sub-tile.
__device__ __forceinline__ float mlp_tile(const MlpW& W,
                                          const float (*__restrict__ feat)[25],
                                          int s, int lane) {
  const int l16 = lane & 15;
  const int kk0 = (lane >= 16) ? 2 : 0;
  const float* row = feat[s * 16 + l16];
  v2f A[6];
#pragma unroll
  for (int kt = 0; kt < 6; ++kt) {
    A[kt].x = row[kt * 4 + kk0];
    A[kt].y = row[kt * 4 + kk0 + 1];
  }
  v8f C[4] = {v8f{}, v8f{}, v8f{}, v8f{}};
#pragma unroll
  for (int kt = 0; kt < 6; ++kt) {
#pragma unroll
    for (int nt = 0; nt < 4; ++nt) {
      C[nt] = __builtin_amdgcn_wmma_f32_16x16x4_f32(
          false, A[kt], false, W.Bf[kt][nt], (short)0, C[nt], false, false);
    }
  }
  // layer 2: relu(h + b1) . w2 ; lane owns column n = nt*16+l16 for 8 rows each
  float acc[8];
#pragma unroll
  for (int r = 0; r < 8; ++r) acc[r] = 0.f;
#pragma unroll
  for (int nt = 0; nt < 4; ++nt) {
#pragma unroll
    for (int r = 0; r < 8; ++r)
      acc[r] += fmaxf(C[nt][r] + W.b1v[nt], 0.f) * W.w2v[nt];
  }
  // butterfly across the 16 lanes sharing each row set (masks 1..8 stay in-half)
#pragma unroll
  for (int m2 = 1; m2 <= 8; m2 <<= 1) {
#pragma unroll
    for (int r = 0; r < 8; ++r) acc[r] += __shfl_xor(acc[r], m2, 32);
  }
  float v = acc[0];
#pragma unroll
  for (int r = 1; r < 8; ++r)
    if ((lane & 7) == r) v = acc[r];
  return v + W.b2s;
}

// ---- kernel A: opacity path, one block per line b ----
// Each wave's LDS staging region is private -> no block barriers in the hot
// loop (same-wave LDS ops are in-order on CDNA5).
__global__ __launch_bounds__(256, 1) void opacity_kernel(
    const float* __restrict__ pts, const float* __restrict__ gt,
    const float2* __restrict__ table,
    const float* __restrict__ w1, const float* __restrict__ b1,
    const float* __restrict__ w2, const float* __restrict__ b2,
    float* __restrict__ sum_out, int P, Levels specs) {
  __shared__ float feat[8][32][25];   // stride 25: gcd(25,64)=1 -> conflict-free
  __shared__ float bsum[8];
  const int b    = blockIdx.x;
  const int lane = threadIdx.x & 31;
  const int wv   = threadIdx.x >> 5;

  const MlpW W = load_mlp(w1, b1, w2, b2, lane);

  float lsum = 0.f;
  for (int base = 0; base < P; base += 256) {
    const int p  = base + wv * 32 + lane;
    const int pc = min(p, P - 1);               // clamp: WMMA needs full EXEC
    const float* xp = pts + ((size_t)b * P + pc) * 3;
    float x01[3] = {(xp[0] + 1.f) * 0.5f, (xp[1] + 1.f) * 0.5f, (xp[2] + 1.f) * 0.5f};
    __builtin_prefetch(pts + ((size_t)b * P + min(pc + 256, P - 1)) * 3, 0, 3);

    encode_point<3>(x01, table, specs, &feat[wv][lane][0]);
#pragma unroll
    for (int s = 0; s < 2; ++s) {
      float o = mlp_tile(W, feat[wv], s, lane);
      int m  = ((lane >= 16) ? 8 : 0) + (lane & 7);
      int pp = base + wv * 32 + s * 16 + m;
      if ((lane & 15) < 8 && pp < P) {
        float cw = (pp < 500) ? (4.f / 3.f) : (2.f / 3.f);
        lsum += fabsf(o - gt[(size_t)b * P + pp]) * cw;
      }
    }
  }
#pragma unroll
  for (int m2 = 16; m2 >= 1; m2 >>= 1) lsum += __shfl_xor(lsum, m2, 32);
  if (lane == 0) bsum[wv] = lsum;
  __syncthreads();
  if (threadIdx.x == 0) {
    float t = 0.f;
#pragma unroll
    for (int i = 0; i < 8; ++i) t += bsum[i];
    sum_out[b] = t;
  }
}

// ---- kernel B: confidence path + per-line loss ----
__global__ __launch_bounds__(256, 1) void conf_kernel(
    const float* __restrict__ line, const float2* __restrict__ table,
    const float* __restrict__ w1, const float* __restrict__ b1,
    const float* __restrict__ w2, const float* __restrict__ b2,
    const float* __restrict__ sum_in, float* __restrict__ lossb,
    int Btot, int P, Levels specs) {
  __shared__ float feat[8][32][25];
  const int lane = threadIdx.x & 31;
  const int wv   = threadIdx.x >> 5;

  const MlpW W = load_mlp(w1, b1, w2, b2, lane);

  const int g  = blockIdx.x * 256 + wv * 32 + lane;
  const int gc = min(g, Btot - 1);
  const float* xp = line + (size_t)gc * 2;
  float x01[2] = {(xp[0] + 1.f) * 0.5f, (xp[1] + 1.f) * 0.5f};
  encode_point<2>(x01, table, specs, &feat[wv][lane][0]);

  const float invP = 1.f / (float)P;
#pragma unroll
  for (int s = 0; s < 2; ++s) {
    float conf = mlp_tile(W, feat[wv], s, lane);
    int m  = ((lane >= 16) ? 8 : 0) + (lane & 7);
    int gm = blockIdx.x * 256 + wv * 32 + s * 16 + m;
    if ((lane & 15) < 8 && gm < Btot) {
      lossb[gm] = expf(-conf) * (sum_in[gm] * invP) + conf;
    }
  }
}

// ---- kernel C: final mean ----
__global__ void reduce_kernel(const float* __restrict__ lossb,
                              float* __restrict__ out, int Btot) {
  __shared__ float sm[256];
  float t = 0.f;
  for (int i = threadIdx.x; i < Btot; i += 256) t += lossb[i];
  sm[threadIdx.x] = t;
  __syncthreads();
  for (int s = 128; s > 0; s >>= 1) {
    if ((int)threadIdx.x < s) sm[threadIdx.x] += sm[threadIdx.x + s];
    __syncthreads();
  }
  if (threadIdx.x == 0) out[0] = sm[0] / (float)Btot;
}

extern "C" void kernel_launch(void* const* d_in, const int* in_sizes, int n_in,
                              void* d_out, int out_size, void* d_ws, size_t ws_size,
                              hipStream_t stream) {
  const float*  line    = (const float*)d_in[0];
  const float*  pts     = (const float*)d_in[1];
  const float*  gt      = (const float*)d_in[2];
  const float2* table_c = (const float2*)d_in[3];
  const float*  w1_c    = (const float*)d_in[4];
  const float*  b1_c    = (const float*)d_in[5];
  const float*  w2_c    = (const float*)d_in[6];
  const float*  b2_c    = (const float*)d_in[7];
  const float2* table_o = (const float2*)d_in[8];
  const float*  w1_o    = (const float*)d_in[9];
  const float*  b1_o    = (const float*)d_in[10];
  const float*  w2_o    = (const float*)d_in[11];
  const float*  b2_o    = (const float*)d_in[12];

  const int B = in_sizes[0] / 2;          // 4096
  const int P = in_sizes[2] / B;          // 1000

  float* sum_b = (float*)d_ws;            // [B]
  float* lossb = sum_b + B;               // [B]
  float* out   = (float*)d_out;

  const Levels L3 = make_levels(3);
  const Levels L2 = make_levels(2);

  opacity_kernel<<<B, 256, 0, stream>>>(pts, gt, table_o, w1_o, b1_o, w2_o, b2_o,
                                        sum_b, P, L3);
  conf_kernel<<<(B + 255) / 256, 256, 0, stream>>>(line, table_c, w1_c, b1_c, w2_c,
                                                   b2_c, sum_b, lossb, B, P, L2);
  reduce_kernel<<<1, 256, 0, stream>>>(lossb, out, B);
}